// QuantumKernelMethod_65481071407575
// MI455X (gfx1250) — compile-verified
//
#include <hip/hip_runtime.h>
#include <hip/hip_bf16.h>
#include <math.h>

// out[i,j] = |cos(a[i,0]-b[j,0])| = |cos(a_i)cos(b_j) + sin(a_i)sin(b_j)|
// (the parameterized block V cancels exactly: U(y)^dag U(x) = Ry(-y)V^dag V Ry(x))
// Rank-2 GEMM, mapped onto v_wmma_f32_16x16x4_f32 (K=4, top 2 K-slices zero).
// Output-bandwidth bound: 4 MB @ 23.3 TB/s ~ 0.18 us; store path kept branch-free.

typedef float v2f __attribute__((ext_vector_type(2)));
typedef float v8f __attribute__((ext_vector_type(8)));

// ---------------- Pass 1: trig precompute (cos/sin of wire-0 angles) --------
__global__ void qkm_trig_kernel(const float* __restrict__ a,
                                const float* __restrict__ b,
                                float* __restrict__ ca, float* __restrict__ sa,
                                float* __restrict__ cb, float* __restrict__ sb,
                                int na, int nb, int stride) {
  int i = blockIdx.x * blockDim.x + threadIdx.x;
  if (i < na) {
    float t = a[i * stride];          // wire 0 of row i
    ca[i] = cosf(t);
    sa[i] = sinf(t);
  } else if (i < na + nb) {
    int j = i - na;
    float t = b[j * stride];
    cb[j] = cosf(t);
    sb[j] = sinf(t);
  }
}

// ---------------- Pass 2: rank-2 GEMM via WMMA, one wave per 16x16 tile -----
// Launched over FULL tiles only -> no bounds checks anywhere in this kernel.
// A-frag (f32 16x4, MxK): lanes 0-15 -> M=lane, VGPR0=K0(cos), VGPR1=K1(sin);
//                         lanes 16-31 -> K2,K3 (zeroed).
// B-frag (f32 4x16, KxN): lanes 0-15 -> N=lane, first K half;
//                         lanes 16-31 -> second K half (zeroed).
// C/D (f32 16x16): VGPR r: lanes 0-15 -> M=r, N=lane;
//                          lanes 16-31 -> M=r+8, N=lane-16.
__global__ void qkm_gram_wmma_kernel(const float* __restrict__ ca,
                                     const float* __restrict__ sa,
                                     const float* __restrict__ cb,
                                     const float* __restrict__ sb,
                                     float* __restrict__ out,
                                     int nb, int tilesN, int totalWaves) {
  const int wave = (blockIdx.x * blockDim.x + threadIdx.x) >> 5;
  const int lane = threadIdx.x & 31;
  if (wave >= totalWaves) return;        // wave-uniform (scalar) guard only

  const int tm = (wave / tilesN) << 4;   // tile row base (M)
  const int tn = (wave % tilesN) << 4;   // tile col base (N)

  v2f afrag = {0.0f, 0.0f};
  v2f bfrag = {0.0f, 0.0f};
  if (lane < 16) {                       // K=0 (cos), K=1 (sin); K=2,3 stay 0
    afrag.x = ca[tm + lane];
    afrag.y = sa[tm + lane];
    bfrag.x = cb[tn + lane];
    bfrag.y = sb[tn + lane];
  }
  // EXEC reconverges here; all-1s at the WMMA as required.

  v8f acc = {};
  // D = A(16x4) * B(4x16) + 0  ->  v_wmma_f32_16x16x4_f32
  acc = __builtin_amdgcn_wmma_f32_16x16x4_f32(
      /*neg_a=*/false, afrag,
      /*neg_b=*/false, bfrag,
      /*c_mod=*/(short)0, acc,
      /*reuse_a=*/false, /*reuse_b=*/false);

  // 32-bit element index (out_size is int, so this cannot overflow):
  // enables global_store_b32 v, voff, s[base] scale_offset addressing.
  const int half = lane >> 4;                  // 0: rows tm..tm+7, 1: tm+8..
  unsigned idx = (unsigned)(tm + (half << 3)) * (unsigned)nb
               + (unsigned)(tn + (lane & 15));
#pragma unroll
  for (int r = 0; r < 8; ++r) {
    out[idx] = __builtin_fabsf(acc[r]);
    idx += (unsigned)nb;
  }
}

// ---------------- Pass 2b: scalar fallback for ragged edge tiles ------------
// Only launched when na%16 || nb%16 (never for 1024x1024).
__global__ void qkm_gram_edge_kernel(const float* __restrict__ ca,
                                     const float* __restrict__ sa,
                                     const float* __restrict__ cb,
                                     const float* __restrict__ sb,
                                     float* __restrict__ out,
                                     int na, int nb, int fullM, int fullN) {
  int t = blockIdx.x * blockDim.x + threadIdx.x;
  // Edge region = rows [fullM, na) over all cols, plus cols [fullN, nb) over
  // rows [0, fullM).
  int bottom = (na - fullM) * nb;
  int m, n;
  if (t < bottom) {
    m = fullM + t / nb;
    n = t % nb;
  } else {
    t -= bottom;
    int w = nb - fullN;
    if (w <= 0) return;
    int total = fullM * w;
    if (t >= total) return;
    m = t / w;
    n = fullN + t % w;
  }
  out[(size_t)m * nb + n] = __builtin_fabsf(ca[m] * cb[n] + sa[m] * sb[n]);
}

extern "C" void kernel_launch(void* const* d_in, const int* in_sizes, int n_in,
                              void* d_out, int out_size, void* d_ws, size_t ws_size,
                              hipStream_t stream) {
  const float* a = (const float*)d_in[0];   // (na, 4) float32
  const float* b = (const float*)d_in[1];   // (nb, 4) float32
  // d_in[2] = params — cancels exactly (V^dag V = I), unused.

  const int NQ = 4;
  const int na = in_sizes[0] / NQ;
  const int nb = in_sizes[1] / NQ;

  float* ca = (float*)d_ws;
  float* sa = ca + na;
  float* cb = sa + na;
  float* sb = cb + nb;

  {
    int total = na + nb;
    int blk = 256;
    int grd = (total + blk - 1) / blk;
    qkm_trig_kernel<<<grd, blk, 0, stream>>>(a, b, ca, sa, cb, sb, na, nb, NQ);
  }

  const int tilesM = na >> 4;                        // full tiles only
  const int tilesN = nb >> 4;
  if (tilesM > 0 && tilesN > 0) {
    const int totalWaves = tilesM * tilesN;          // 4096 for 1024x1024
    const int wavesPerBlock = 8;                     // 256 threads, wave32
    const int blk = wavesPerBlock * 32;
    const int grd = (totalWaves + wavesPerBlock - 1) / wavesPerBlock;
    qkm_gram_wmma_kernel<<<grd, blk, 0, stream>>>(ca, sa, cb, sb,
                                                  (float*)d_out, nb, tilesN,
                                                  totalWaves);
  }

  const int fullM = tilesM << 4;
  const int fullN = tilesN << 4;
  if (fullM < na || fullN < nb) {                    // ragged edges (not 1024)
    int edgeElems = (na - fullM) * nb + fullM * (nb - fullN);
    if (edgeElems > 0) {
      int blk = 256;
      int grd = (edgeElems + blk - 1) / blk;
      qkm_gram_edge_kernel<<<grd, blk, 0, stream>>>(ca, sa, cb, sb,
                                                    (float*)d_out,
                                                    na, nb, fullM, fullN);
    }
  }
}